// AJBSMamba_50130858279434
// MI455X (gfx1250) — compile-verified
//
#include <hip/hip_runtime.h>
#include <math.h>

typedef float v2f __attribute__((ext_vector_type(2)));
typedef float v8f __attribute__((ext_vector_type(8)));

#define B_SZ   64
#define L_SZ   512
#define DMODEL 96
#define DSTATE 4
#define DINNER 192
#define DTRANK 6
#define NTOK   (B_SZ * L_SZ)   // 32768 tokens

// ---------------- workspace layout (float offsets) ----------------
#define F_XV    ((size_t)0)                          // NTOK*DINNER  (reused as y_raw)
#define F_Z     (F_XV    + (size_t)NTOK * DINNER)    // NTOK*DINNER  (reused as y_gated)
#define F_US    (F_Z     + (size_t)NTOK * DINNER)
#define F_DELTA (F_US    + (size_t)NTOK * DINNER)
#define F_ZS    (F_DELTA + (size_t)NTOK * DINNER)
#define F_XDBL  (F_ZS    + (size_t)NTOK * DINNER)    // NTOK*16 (stride 16, cols 0..13 valid)
#define F_BS    (F_XDBL  + (size_t)NTOK * 16)
#define F_CS    (F_BS    + (size_t)NTOK * 4)
#define F_S     (F_CS    + (size_t)NTOK * 4)
#define F_IC    (F_S     + (size_t)NTOK)             // int
#define F_MASK  (F_IC    + (size_t)NTOK)             // int

// ---------------- fp32 WMMA 16x16 tile helper ----------------
// Computes Out[16x16] tile of  A(MxK) * B(NxK)^T  with V_WMMA_F32_16X16X4_F32.
// A 16x4 layout: lanes 0-15 hold (M=lane, K=k0..k0+1), lanes 16-31 (M=lane-16, K=k0+2..k0+3).
// B rows beyond NVALID are zero-filled (branchless per-lane select; EXEC uniform at WMMA).
template <int K, int NVALID>
__device__ __forceinline__ v8f wmma_f32_tile(const float* __restrict__ A, int lda,
                                             const float* __restrict__ Bw, int ldb) {
  const int lane = threadIdx.x & 31;
  const int r16  = lane & 15;
  const int klo  = (lane >> 4) << 1;
  v8f acc = {};
#pragma unroll
  for (int k0 = 0; k0 < K; k0 += 4) {
    v2f a;
    a.x = A[r16 * lda + k0 + klo];
    a.y = A[r16 * lda + k0 + klo + 1];
    v2f b;
    const bool ok = (NVALID >= 16) || (r16 < NVALID);
    b.x = ok ? Bw[r16 * ldb + k0 + klo]     : 0.0f;
    b.y = ok ? Bw[r16 * ldb + k0 + klo + 1] : 0.0f;
    acc = __builtin_amdgcn_wmma_f32_16x16x4_f32(false, a, false, b, (short)0, acc,
                                                false, false);
  }
  return acc;
}

// ---------------- GEMM1: xz = x @ in_proj_w^T, split into xv | z ----------------
__global__ __launch_bounds__(128) void k_gemm_inproj(const float* __restrict__ x,
                                                     const float* __restrict__ w,
                                                     float* __restrict__ xv,
                                                     float* __restrict__ z) {
  const int wave  = threadIdx.x >> 5;
  const int lane  = threadIdx.x & 31;
  const int mtile = blockIdx.x * 4 + wave;
  const int ntile = blockIdx.y;
  v8f acc = wmma_f32_tile<DMODEL, 16>(x + (size_t)mtile * 16 * DMODEL, DMODEL,
                                      w + (size_t)ntile * 16 * DMODEL, DMODEL);
  const int n    = lane & 15;
  const int half = lane >> 4;
  const int ncol = ntile * 16 + n;
#pragma unroll
  for (int r = 0; r < 8; ++r) {
    const int row = mtile * 16 + r + 8 * half;
    const float v = acc[r];
    if (ncol < DINNER) xv[(size_t)row * DINNER + ncol] = v;
    else               z[(size_t)row * DINNER + (ncol - DINNER)] = v;
  }
}

// ---------------- GEMM2: x_dbl = xv @ x_proj_w^T (14 cols, stride-16 store) ----------------
__global__ __launch_bounds__(128) void k_gemm_xproj(const float* __restrict__ xv,
                                                    const float* __restrict__ w,
                                                    float* __restrict__ xdbl) {
  const int wave  = threadIdx.x >> 5;
  const int lane  = threadIdx.x & 31;
  const int mtile = blockIdx.x * 4 + wave;
  v8f acc = wmma_f32_tile<DINNER, 14>(xv + (size_t)mtile * 16 * DINNER, DINNER,
                                      w, DINNER);
  const int n    = lane & 15;
  const int half = lane >> 4;
  if (n < DTRANK + 2 * DSTATE) {
#pragma unroll
    for (int r = 0; r < 8; ++r)
      xdbl[(size_t)(mtile * 16 + r + 8 * half) * 16 + n] = acc[r];
  }
}

// ---------------- GEMM3: out = y_gated @ out_proj_w^T ----------------
__global__ __launch_bounds__(128) void k_gemm_outproj(const float* __restrict__ yg,
                                                      const float* __restrict__ w,
                                                      float* __restrict__ out) {
  const int wave  = threadIdx.x >> 5;
  const int lane  = threadIdx.x & 31;
  const int mtile = blockIdx.x * 4 + wave;
  const int ntile = blockIdx.y;
  v8f acc = wmma_f32_tile<DINNER, 16>(yg + (size_t)mtile * 16 * DINNER, DINNER,
                                      w + (size_t)ntile * 16 * DINNER, DINNER);
  const int n    = lane & 15;
  const int half = lane >> 4;
#pragma unroll
  for (int r = 0; r < 8; ++r)
    out[(size_t)(mtile * 16 + r + 8 * half) * DMODEL + ntile * 16 + n] = acc[r];
}

// ---------------- s = xv.mean(-1): one wave32 per row ----------------
__global__ __launch_bounds__(256) void k_row_mean(const float* __restrict__ xv,
                                                  float* __restrict__ s) {
  const int wave = threadIdx.x >> 5;
  const int lane = threadIdx.x & 31;
  const int row  = blockIdx.x * 8 + wave;
  float acc = 0.0f;
  for (int k = lane; k < DINNER; k += 32) acc += xv[(size_t)row * DINNER + k];
#pragma unroll
  for (int off = 16; off > 0; off >>= 1) acc += __shfl_down(acc, off, 32);
  if (lane == 0) s[row] = acc * (1.0f / DINNER);
}

// ---------------- curvature -> steps -> sequential index build ----------------
__global__ __launch_bounds__(L_SZ) void k_build_steps(const float* __restrict__ s,
                                                      int* __restrict__ ic,
                                                      int* __restrict__ mask) {
  __shared__ float ssh[L_SZ];
  __shared__ float csh[L_SZ];
  __shared__ float rmin[L_SZ];
  __shared__ float rmax[L_SZ];
  __shared__ int   stsh[L_SZ];
  const int b = blockIdx.x, t = threadIdx.x;
  ssh[t] = s[b * L_SZ + t];
  __syncthreads();
  float c = 0.0f;
  if (t < L_SZ - 1) {
    float dm1 = 0.0f, dp1 = 0.0f;
    if (t >= 1)          { float d = ssh[t] - ssh[t - 1];     dm1 = d * d; }
    {                      float d = ssh[t + 1] - ssh[t];     c   = d * d; }
    if (t + 2 <= L_SZ - 1) { float d = ssh[t + 2] - ssh[t + 1]; dp1 = d * d; }
    c = (dm1 + c + dp1) * (1.0f / 3.0f);
  }
  if (isnan(c)) c = 0.0f;
  else if (isinf(c)) c = (c > 0.0f) ? 1.0f : 0.0f;
  csh[t] = c; rmin[t] = c; rmax[t] = c;
  __syncthreads();
  for (int off = L_SZ / 2; off > 0; off >>= 1) {
    if (t < off) {
      rmin[t] = fminf(rmin[t], rmin[t + off]);
      rmax[t] = fmaxf(rmax[t], rmax[t + off]);
    }
    __syncthreads();
  }
  const float cmin = rmin[0], cmax = rmax[0];
  const float rng  = cmax - cmin;
  float cn = (rng > 0.0f) ? (csh[t] - cmin) / (rng + 1e-6f) : 0.0f;
  cn = fminf(fmaxf(cn, 0.0f), 1.0f);
  const float sig = 1.0f / (1.0f + __expf(-5.0f * (cn - 0.5f)));
  float stf = 1.0f + 2.0f * (1.0f - sig);
  stf = fminf(fmaxf(stf, 1.0f), 3.0f);
  int st = (int)rintf(stf);
  if (st < 1) st = 1;
  stsh[t] = st;
  __syncthreads();
  if (t == 0) {
    int idx = 0;
    for (int i = 0; i < L_SZ; ++i) {
      const bool in = (idx < L_SZ);
      mask[b * L_SZ + i] = in ? 1 : 0;
      ic[b * L_SZ + i]   = in ? idx : (L_SZ - 1);
      int cl = idx; if (cl < 0) cl = 0; if (cl > L_SZ - 1) cl = L_SZ - 1;
      int stp = stsh[cl]; if (stp < 1) stp = 1;
      if (in) idx += stp;
    }
  }
}

__device__ __forceinline__ float softplus_f(float x) {
  return (x > 20.0f) ? x : log1pf(__expf(x));
}

// ---------------- gather + dt_proj + delta ----------------
__global__ __launch_bounds__(DINNER) void k_gather(const float* __restrict__ xv,
                                                   const float* __restrict__ z,
                                                   const float* __restrict__ xdbl,
                                                   const int* __restrict__ ic,
                                                   const int* __restrict__ mask,
                                                   const float* __restrict__ dtw,
                                                   const float* __restrict__ dtb,
                                                   float* __restrict__ us,
                                                   float* __restrict__ delta,
                                                   float* __restrict__ zs,
                                                   float* __restrict__ Bs,
                                                   float* __restrict__ Cs) {
  __shared__ float xd[16];
  const int tok  = blockIdx.x;
  const int b    = tok >> 9;
  const int icv  = ic[tok];
  const float fm = (float)mask[tok];
  const int srow = (b << 9) + icv;
  const int d    = threadIdx.x;
  if (d < 16) xd[d] = xdbl[(size_t)srow * 16 + d];
  __syncthreads();
  const float u  = fm * xv[(size_t)srow * DINNER + d];
  const float zz = fm * z[(size_t)srow * DINNER + d];
  float dtv = 0.0f;
#pragma unroll
  for (int r = 0; r < DTRANK; ++r) dtv += dtw[d * DTRANK + r] * xd[r];
  dtv *= fm;
  us[(size_t)tok * DINNER + d]    = u;
  zs[(size_t)tok * DINNER + d]    = zz;
  delta[(size_t)tok * DINNER + d] = softplus_f(dtv) + dtb[d];
  if (d < DSTATE) {
    Bs[(size_t)tok * DSTATE + d] = fm * xd[DTRANK + d];
    Cs[(size_t)tok * DSTATE + d] = fm * xd[DTRANK + DSTATE + d];
  }
}

// ---------------- selective scan: one lane per (batch, channel) ----------------
__global__ __launch_bounds__(DINNER) void k_scan(const float* __restrict__ us,
                                                 const float* __restrict__ delta,
                                                 const float* __restrict__ Bs,
                                                 const float* __restrict__ Cs,
                                                 const float* __restrict__ A_log,
                                                 const float* __restrict__ Dp,
                                                 float* __restrict__ yraw) {
  const int b = blockIdx.x, d = threadIdx.x;
  float Ad[DSTATE];
#pragma unroll
  for (int n = 0; n < DSTATE; ++n) Ad[n] = -__expf(A_log[d * DSTATE + n]);
  const float Dv = Dp[d];
  float h[DSTATE] = {0.0f, 0.0f, 0.0f, 0.0f};
  for (int t = 0; t < L_SZ; ++t) {
    const size_t tok = (size_t)b * L_SZ + t;
    const float dl = delta[tok * DINNER + d];
    const float u  = us[tok * DINNER + d];
    const float du = dl * u;
    float y = Dv * u;
#pragma unroll
    for (int n = 0; n < DSTATE; ++n) {
      h[n] = __expf(dl * Ad[n]) * h[n] + du * Bs[tok * DSTATE + n];
      y += h[n] * Cs[tok * DSTATE + n];
    }
    yraw[tok * DINNER + d] = y;
  }
}

// ---------------- LayerNorm + SiLU gate ----------------
__global__ __launch_bounds__(DINNER) void k_ln_gate(const float* __restrict__ yraw,
                                                    const float* __restrict__ zs,
                                                    const float* __restrict__ lw,
                                                    const float* __restrict__ lb,
                                                    float* __restrict__ yg) {
  __shared__ float red[256];
  const int tok = blockIdx.x, d = threadIdx.x;
  const float v = yraw[(size_t)tok * DINNER + d];
  red[d] = v;
  if (d < 64) red[DINNER + d] = 0.0f;
  __syncthreads();
  for (int off = 128; off > 0; off >>= 1) {
    if (d < off) red[d] += red[d + off];
    __syncthreads();
  }
  const float mu = red[0] * (1.0f / DINNER);
  __syncthreads();
  const float dv = v - mu;
  red[d] = dv * dv;
  if (d < 64) red[DINNER + d] = 0.0f;
  __syncthreads();
  for (int off = 128; off > 0; off >>= 1) {
    if (d < off) red[d] += red[d + off];
    __syncthreads();
  }
  const float var = red[0] * (1.0f / DINNER);
  const float yl  = dv * rsqrtf(var + 1e-5f) * lw[d] + lb[d];
  const float zz  = zs[(size_t)tok * DINNER + d];
  const float sg  = 1.0f / (1.0f + __expf(-zz));
  yg[(size_t)tok * DINNER + d] = yl * (zz * sg);
}

// ---------------- launch ----------------
extern "C" void kernel_launch(void* const* d_in, const int* in_sizes, int n_in,
                              void* d_out, int out_size, void* d_ws, size_t ws_size,
                              hipStream_t stream) {
  const float* x        = (const float*)d_in[0];
  const float* in_proj  = (const float*)d_in[1];
  const float* x_proj   = (const float*)d_in[2];
  const float* dt_proj  = (const float*)d_in[3];
  const float* dt_b     = (const float*)d_in[4];
  const float* A_log    = (const float*)d_in[5];
  const float* D_param  = (const float*)d_in[6];
  const float* ln_w     = (const float*)d_in[7];
  const float* ln_b     = (const float*)d_in[8];
  const float* out_proj = (const float*)d_in[9];
  float* out = (float*)d_out;

  float* ws = (float*)d_ws;
  float* xv    = ws + F_XV;
  float* z     = ws + F_Z;
  float* us    = ws + F_US;
  float* delta = ws + F_DELTA;
  float* zs    = ws + F_ZS;
  float* xdbl  = ws + F_XDBL;
  float* Bs    = ws + F_BS;
  float* Cs    = ws + F_CS;
  float* s     = ws + F_S;
  int*   ic    = (int*)(ws + F_IC);
  int*   mask  = (int*)(ws + F_MASK);
  float* yraw  = xv;   // xv dead after gather
  float* yg    = z;    // z dead after gather

  const int mblk = (NTOK / 16) / 4;  // 512 blocks of 4 waves (M tiles)

  k_gemm_inproj<<<dim3(mblk, (2 * DINNER) / 16), 128, 0, stream>>>(x, in_proj, xv, z);
  k_row_mean<<<NTOK / 8, 256, 0, stream>>>(xv, s);
  k_gemm_xproj<<<dim3(mblk, 1), 128, 0, stream>>>(xv, x_proj, xdbl);
  k_build_steps<<<B_SZ, L_SZ, 0, stream>>>(s, ic, mask);
  k_gather<<<NTOK, DINNER, 0, stream>>>(xv, z, xdbl, ic, mask, dt_proj, dt_b,
                                        us, delta, zs, Bs, Cs);
  k_scan<<<B_SZ, DINNER, 0, stream>>>(us, delta, Bs, Cs, A_log, D_param, yraw);
  k_ln_gate<<<NTOK, DINNER, 0, stream>>>(yraw, zs, ln_w, ln_b, yg);
  k_gemm_outproj<<<dim3(mblk, DMODEL / 16), 128, 0, stream>>>(yg, out_proj, out);
}